// NetTGCNBasicFFT_13288628814249
// MI455X (gfx1250) — compile-verified
//
#include <hip/hip_runtime.h>
#include <math.h>

// Problem constants (from reference)
#define NN   100000
#define EE   3200000
#define TT   15
#define KCH  6
#define FOUT 10
#define NC   6
#define FRQ  8
#define KP   96    // padded inner dim: 6*15=90 -> 96 (multiple of 4 and 16)
#define JP   160   // padded output dim: 10*15=150 -> 160

typedef float v2f __attribute__((ext_vector_type(2)));
typedef float v8f __attribute__((ext_vector_type(8)));

// workspace layout in floats
#define TX_OFF   ((size_t)0)
#define TX_SZ    ((size_t)NN * KP)          // 9,600,000 floats
#define DEG_OFF  (TX_OFF + TX_SZ)
#define DEG_SZ   ((size_t)NN)
#define BM_OFF   (DEG_OFF + DEG_SZ)
#define BM_SZ    ((size_t)KP * JP)          // 15,360 floats
#define LACC_OFF (BM_OFF + BM_SZ)
#define LACC_SZ  ((size_t)NC)
#define WS_TOTAL (LACC_OFF + LACC_SZ)

__global__ void zero_kernel(float* __restrict__ p, size_t n) {
    size_t i = (size_t)blockIdx.x * blockDim.x + threadIdx.x;
    size_t stride = (size_t)gridDim.x * blockDim.x;
    for (; i < n; i += stride) p[i] = 0.0f;
}

// deg[row] += 1 over all edges
__global__ void deg_kernel(const int* __restrict__ ei, float* __restrict__ deg) {
    int e = blockIdx.x * blockDim.x + threadIdx.x;
    if (e < EE) atomicAdd(&deg[ei[e]], 1.0f);
}

// in-place deg -> d^{-1/2}
__global__ void dinv_kernel(float* __restrict__ deg) {
    int n = blockIdx.x * blockDim.x + threadIdx.x;
    if (n < NN) {
        float d = deg[n];
        deg[n] = (d > 0.0f) ? rsqrtf(d) : 0.0f;
    }
}

// TX[n, 0:15] = x[n, 0, 0:15]
__global__ void copyx_kernel(const float* __restrict__ x, float* __restrict__ TX) {
    int i = blockIdx.x * blockDim.x + threadIdx.x;
    if (i < NN * TT) {
        int n = i / TT, t = i % TT;
        TX[(size_t)n * KP + t] = x[i];
    }
}

// TX[row, k*15+t] += scale * (-(dinv[row]*dinv[col])) * TX[col, (k-1)*15+t]
__global__ void prop_kernel(const int* __restrict__ ei, const float* __restrict__ dinv,
                            float* __restrict__ TX, int k, float scale) {
    int e = blockIdx.x * blockDim.x + threadIdx.x;
    if (e >= EE) return;
    int r = ei[e];
    int c = ei[EE + e];
    float w = -scale * dinv[r] * dinv[c];
    const float* __restrict__ src = TX + (size_t)c * KP + (size_t)(k - 1) * TT;
    float* __restrict__ dst = TX + (size_t)r * KP + (size_t)k * TT;
#pragma unroll
    for (int t = 0; t < TT; ++t) atomicAdd(dst + t, w * src[t]);
}

// Chebyshev: tx_k -= tx_{k-2}   (the 2x is already applied in prop scale)
__global__ void cheb_sub_kernel(float* __restrict__ TX, int k) {
    int i = blockIdx.x * blockDim.x + threadIdx.x;
    if (i < NN * TT) {
        int n = i / TT, t = i % TT;
        TX[(size_t)n * KP + (size_t)k * TT + t] -=
            TX[(size_t)n * KP + (size_t)(k - 2) * TT + t];
    }
}

// Fold rfft -> per-bin complex filter -> irfft into one real matrix:
// B[k*15+ti, fo*15+t] = (1/15) [ Wr[0] + 2*sum_{f=1..7} (Wr[f] cos th - Wi[f] sin th) ],
// th = 2*pi*f*(t-ti)/15
__global__ void buildB_kernel(const float* __restrict__ w_real,
                              const float* __restrict__ w_imag,
                              float* __restrict__ Bm) {
    int id = blockIdx.x * blockDim.x + threadIdx.x;
    if (id >= KCH * FOUT * TT * TT) return;
    int k   = id / (FOUT * TT * TT);
    int rem = id % (FOUT * TT * TT);
    int fo  = rem / (TT * TT);
    int tt2 = rem % (TT * TT);
    int t = tt2 / TT, ti = tt2 % TT;
    const float* wr = w_real + ((size_t)k * FOUT + fo) * FRQ;  // [K,1,FOUT,FR]
    const float* wi = w_imag + ((size_t)k * FOUT + fo) * FRQ;
    float acc  = wr[0];
    float base = 6.28318530717958647692f * (float)(t - ti) * (1.0f / 15.0f);
#pragma unroll
    for (int f = 1; f < FRQ; ++f) {
        float th = base * (float)f;
        acc += 2.0f * (wr[f] * cosf(th) - wi[f] * sinf(th));
    }
    Bm[(size_t)(k * TT + ti) * JP + (size_t)(fo * TT + t)] = acc * (1.0f / 15.0f);
}

// WMMA GEMM:  OUT[N,150] = TX[N,96] @ B[96,160]  fused with
// relu(out + conv_bias) and fc_w dot-products -> per-class partial sums.
// One wave owns one 16x16 output tile; K-loop = 24x V_WMMA_F32_16X16X4_F32.
__global__ __launch_bounds__(256) void gemm_fc_kernel(const float* __restrict__ TX,
                                                      const float* __restrict__ Bm,
                                                      const float* __restrict__ conv_bias,
                                                      const float* __restrict__ fc_w,
                                                      float* __restrict__ lacc) {
    const int lane = threadIdx.x & 31;
    const int wave = threadIdx.x >> 5;
    const int gw   = blockIdx.x * 8 + wave;       // global wave id
    const int NROW = NN / 16;                     // 6250 row tiles (exact)
    const int NCOL = JP / 16;                     // 10 col tiles
    float accv[NC] = {0.f, 0.f, 0.f, 0.f, 0.f, 0.f};

    if (gw < NROW * NCOL) {
        const int rt = gw / NCOL;
        const int ct = gw % NCOL;
        const int node_base = rt * 16;
        const int jbase     = ct * 16;
        const int ln   = lane & 15;               // M (for A) / N (for B,C,D)
        const int half = lane >> 4;
        const int koff = half * 2;                // K sub-offset for this lane group

        v8f d = {0.f, 0.f, 0.f, 0.f, 0.f, 0.f, 0.f, 0.f};
        const float* __restrict__ arow = TX + (size_t)(node_base + ln) * KP + koff;
        for (int kk0 = 0; kk0 < KP; kk0 += 4) {
            // A 16x4 f32: lane m=ln, K = kk0 + koff + {0,1}  (8-byte aligned)
            v2f a = *reinterpret_cast<const v2f*>(arow + kk0);
            // B 4x16 f32: col n=ln, K = kk0 + koff + {0,1}
            v2f b;
            b.x = Bm[(size_t)(kk0 + koff + 0) * JP + jbase + ln];
            b.y = Bm[(size_t)(kk0 + koff + 1) * JP + jbase + ln];
            d = __builtin_amdgcn_wmma_f32_16x16x4_f32(
                    false, a, false, b, (short)0, d, false, false);
        }

        // Epilogue: relu(out + conv_bias[fo]) then fc_w dot into class partials.
        const int j = jbase + ln;                 // flattened fo*15+t
        if (j < FOUT * TT) {
            const int fo = j / TT;
            const float cb = conv_bias[fo];
#pragma unroll
            for (int r = 0; r < 8; ++r) {
                const int node = node_base + half * 8 + r;   // D row mapping
                float h = d[r] + cb;
                h = (h > 0.0f) ? h : 0.0f;
                const size_t base = (size_t)node * (FOUT * TT) + (size_t)j;
#pragma unroll
                for (int cc = 0; cc < NC; ++cc)
                    accv[cc] += fc_w[(size_t)cc * ((size_t)NN * FOUT * TT) + base] * h;
            }
        }
    }

    // wave32 reduction, then one atomic per class per wave
#pragma unroll
    for (int off = 16; off > 0; off >>= 1) {
#pragma unroll
        for (int cc = 0; cc < NC; ++cc)
            accv[cc] += __shfl_down(accv[cc], off, 32);
    }
    if (lane == 0) {
#pragma unroll
        for (int cc = 0; cc < NC; ++cc) atomicAdd(&lacc[cc], accv[cc]);
    }
}

__global__ void finish_kernel(const float* __restrict__ lacc,
                              const float* __restrict__ fc_b,
                              float* __restrict__ out) {
    if (threadIdx.x != 0 || blockIdx.x != 0) return;
    float l[NC];
    float mx = -3.4e38f;
#pragma unroll
    for (int c = 0; c < NC; ++c) { l[c] = lacc[c] + fc_b[c]; mx = fmaxf(mx, l[c]); }
    float s = 0.0f;
#pragma unroll
    for (int c = 0; c < NC; ++c) s += expf(l[c] - mx);
    float lse = mx + logf(s);
#pragma unroll
    for (int c = 0; c < NC; ++c) out[c] = l[c] - lse;
}

extern "C" void kernel_launch(void* const* d_in, const int* in_sizes, int n_in,
                              void* d_out, int out_size, void* d_ws, size_t ws_size,
                              hipStream_t stream) {
    const float* x         = (const float*)d_in[0];
    const float* w_real    = (const float*)d_in[1];
    const float* w_imag    = (const float*)d_in[2];
    const float* conv_bias = (const float*)d_in[3];
    const float* fc_w      = (const float*)d_in[4];
    const float* fc_b      = (const float*)d_in[5];
    const int*   ei        = (const int*)d_in[6];   // [2,E]: rows then cols

    float* ws   = (float*)d_ws;
    float* TX   = ws + TX_OFF;
    float* DEG  = ws + DEG_OFF;
    float* BM   = ws + BM_OFF;
    float* LACC = ws + LACC_OFF;
    float* out  = (float*)d_out;

    zero_kernel<<<4096, 256, 0, stream>>>(ws, WS_TOTAL);
    buildB_kernel<<<(KCH * FOUT * TT * TT + 255) / 256, 256, 0, stream>>>(w_real, w_imag, BM);
    deg_kernel<<<(EE + 255) / 256, 256, 0, stream>>>(ei, DEG);
    dinv_kernel<<<(NN + 255) / 256, 256, 0, stream>>>(DEG);
    copyx_kernel<<<(NN * TT + 255) / 256, 256, 0, stream>>>(x, TX);

    for (int k = 1; k < KCH; ++k) {
        prop_kernel<<<(EE + 255) / 256, 256, 0, stream>>>(ei, DEG, TX, k,
                                                          (k == 1) ? 1.0f : 2.0f);
        if (k >= 2)
            cheb_sub_kernel<<<(NN * TT + 255) / 256, 256, 0, stream>>>(TX, k);
    }

    const int total_waves = (NN / 16) * (JP / 16);  // 62,500
    gemm_fc_kernel<<<(total_waves + 7) / 8, 256, 0, stream>>>(TX, BM, conv_bias, fc_w, LACC);
    finish_kernel<<<1, 32, 0, stream>>>(LACC, fc_b, out);
}